// Decoder_8950711845590
// MI455X (gfx1250) — compile-verified
//
#include <hip/hip_runtime.h>

// ---------------------------------------------------------------------------
// PointNet++ FP decoder for MI455X (gfx1250).
// GEMMs run through V_WMMA_F32_16X16X4_F32 (exact fp32 matrix path), with
// 64x32 register blocking per wave (8 WMMAs per K-step, independent
// accumulators to hide WMMA->WMMA RAW hazards).
// kNN / BN / interpolation are lightweight VALU kernels; the whole problem
// (~150MB) is L2-resident on a 192MB-L2, 23.3TB/s part.
// ---------------------------------------------------------------------------

typedef __attribute__((ext_vector_type(2))) float v2f;
typedef __attribute__((ext_vector_type(8))) float v8f;

#define NB 16
#define N0T 65536
#define N1T 16384
#define N2T 4096
#define N3T 1024
#define N0B 4096
#define N1B 1024
#define N2B 256
#define N3B 64

// ---------------------------------------------------------------------------
// GEMM: C[M,N(logical)] = A[M,K] * W[K,N] + bias, optional ReLU.
// Block tile: 256 rows x 64 cols, 8 waves; each wave: 64x32 (4 M-tiles x 2
// N-tiles) of 16x16 WMMA tiles.
// A frag (16x4 fp32): lanes 0-15 -> M=lane, K={k0,k0+1}; lanes 16-31 -> K={k0+2,k0+3}.
// B frag (4x16 fp32): lanes 0-15 -> N=lane, K={k0,k0+1}; lanes 16-31 -> K={k0+2,k0+3}.
// C/D frag: VGPR r -> row r (lanes 0-15) / row r+8 (lanes 16-31), col = lane&15.
// Requires: M % 256 == 0, K % 4 == 0, N % 32 == 0 (pad W); stores col < nstore.
// ---------------------------------------------------------------------------
__global__ __launch_bounds__(256) void gemm_wmma_f32(
    const float* __restrict__ A, int lda,
    const float* __restrict__ W, int ldw,
    const float* __restrict__ bias,
    float* __restrict__ C, int ldc,
    int M, int N, int K, int nstore, int relu)
{
    const int wave = threadIdx.x >> 5;
    const int lane = threadIdx.x & 31;
    const int m0 = blockIdx.x * 256 + (wave & 3) * 64;   // 4 waves over M
    const int n0 = blockIdx.y * 64 + (wave >> 2) * 32;   // 2 waves over N
    if (n0 >= N) return;                 // wave-uniform: EXEC stays all-1s
    const int half = lane >> 4;          // 0: lanes 0-15, 1: lanes 16-31
    const int l    = lane & 15;

    // warm L2/WGP$ for the weight panel shared by all waves (global_prefetch)
    __builtin_prefetch(&W[n0 + l], 0, 1);

    const float* arow0 = A + (size_t)(m0 +  0 + l) * lda;
    const float* arow1 = A + (size_t)(m0 + 16 + l) * lda;
    const float* arow2 = A + (size_t)(m0 + 32 + l) * lda;
    const float* arow3 = A + (size_t)(m0 + 48 + l) * lda;
    const float* wcol0 = W + n0 + l;
    const float* wcol1 = W + n0 + 16 + l;

    v8f acc00 = {}, acc01 = {}, acc10 = {}, acc11 = {};
    v8f acc20 = {}, acc21 = {}, acc30 = {}, acc31 = {};

    for (int k0 = 0; k0 < K; k0 += 4) {
        const int ka = k0 + half * 2;
        v2f a0, a1, a2, a3, b0, b1;
        a0.x = arow0[ka]; a0.y = arow0[ka + 1];
        a1.x = arow1[ka]; a1.y = arow1[ka + 1];
        a2.x = arow2[ka]; a2.y = arow2[ka + 1];
        a3.x = arow3[ka]; a3.y = arow3[ka + 1];
        b0.x = wcol0[(size_t)ka * ldw]; b0.y = wcol0[(size_t)(ka + 1) * ldw];
        b1.x = wcol1[(size_t)ka * ldw]; b1.y = wcol1[(size_t)(ka + 1) * ldw];
        acc00 = __builtin_amdgcn_wmma_f32_16x16x4_f32(false, a0, false, b0, (short)0, acc00, false, false);
        acc10 = __builtin_amdgcn_wmma_f32_16x16x4_f32(false, a1, false, b0, (short)0, acc10, false, false);
        acc20 = __builtin_amdgcn_wmma_f32_16x16x4_f32(false, a2, false, b0, (short)0, acc20, false, false);
        acc30 = __builtin_amdgcn_wmma_f32_16x16x4_f32(false, a3, false, b0, (short)0, acc30, false, false);
        acc01 = __builtin_amdgcn_wmma_f32_16x16x4_f32(false, a0, false, b1, (short)0, acc01, false, false);
        acc11 = __builtin_amdgcn_wmma_f32_16x16x4_f32(false, a1, false, b1, (short)0, acc11, false, false);
        acc21 = __builtin_amdgcn_wmma_f32_16x16x4_f32(false, a2, false, b1, (short)0, acc21, false, false);
        acc31 = __builtin_amdgcn_wmma_f32_16x16x4_f32(false, a3, false, b1, (short)0, acc31, false, false);
    }

    const int  col0 = n0 + l;
    const int  col1 = n0 + 16 + l;
    const bool s0 = col0 < nstore;
    const bool s1 = col1 < nstore;
    const float bv0 = s0 ? bias[col0] : 0.0f;
    const float bv1 = s1 ? bias[col1] : 0.0f;

    v8f accs0[4] = {acc00, acc10, acc20, acc30};
    v8f accs1[4] = {acc01, acc11, acc21, acc31};
    #pragma unroll
    for (int t = 0; t < 4; ++t) {
        #pragma unroll
        for (int r = 0; r < 8; ++r) {
            const int row = m0 + t * 16 + r + half * 8;
            if (s0) {
                float v = accs0[t][r] + bv0;
                if (relu) v = fmaxf(v, 0.0f);
                C[(size_t)row * ldc + col0] = v;
            }
            if (s1) {
                float v = accs1[t][r] + bv1;
                if (relu) v = fmaxf(v, 0.0f);
                C[(size_t)row * ldc + col1] = v;
            }
        }
    }
}

// ---------------------------------------------------------------------------
// kNN, k = 1: one thread per query, scan candidates of its batch.
// ---------------------------------------------------------------------------
__global__ void knn1_kernel(const float* __restrict__ posq,
                            const float* __restrict__ posc,
                            int nq_per_b, int nc_per_b, int nq_total,
                            int* __restrict__ idx)
{
    const int q = blockIdx.x * blockDim.x + threadIdx.x;
    if (q >= nq_total) return;
    const int b = q / nq_per_b;
    const float qx = posq[3 * (size_t)q + 0];
    const float qy = posq[3 * (size_t)q + 1];
    const float qz = posq[3 * (size_t)q + 2];
    const float* cp = posc + (size_t)b * nc_per_b * 3;
    float best = 3.4e38f; int bi = 0;
    for (int j = 0; j < nc_per_b; ++j) {
        const float dx = qx - cp[3 * j + 0];
        const float dy = qy - cp[3 * j + 1];
        const float dz = qz - cp[3 * j + 2];
        const float d = dx * dx + dy * dy + dz * dz;
        if (d < best) { best = d; bi = j; }
    }
    idx[q] = b * nc_per_b + bi;
}

// ---------------------------------------------------------------------------
// kNN, k = 3: stable 3-smallest insertion (matches top_k tie ordering),
// writes global candidate indices and inverse-squared-distance weights.
// ---------------------------------------------------------------------------
__global__ void knn3_kernel(const float* __restrict__ posq,
                            const float* __restrict__ posc,
                            int nq_per_b, int nc_per_b, int nq_total,
                            int* __restrict__ idx, float* __restrict__ wgt)
{
    const int q = blockIdx.x * blockDim.x + threadIdx.x;
    if (q >= nq_total) return;
    const int b = q / nq_per_b;
    const float qx = posq[3 * (size_t)q + 0];
    const float qy = posq[3 * (size_t)q + 1];
    const float qz = posq[3 * (size_t)q + 2];
    const float* cp = posc + (size_t)b * nc_per_b * 3;
    float d0 = 3.4e38f, d1 = 3.4e38f, d2 = 3.4e38f;
    int   i0 = 0,       i1 = 0,       i2 = 0;
    for (int j = 0; j < nc_per_b; ++j) {
        const float dx = qx - cp[3 * j + 0];
        const float dy = qy - cp[3 * j + 1];
        const float dz = qz - cp[3 * j + 2];
        const float d = dx * dx + dy * dy + dz * dz;
        if (d < d0)      { d2 = d1; i2 = i1; d1 = d0; i1 = i0; d0 = d; i0 = j; }
        else if (d < d1) { d2 = d1; i2 = i1; d1 = d;  i1 = j; }
        else if (d < d2) { d2 = d;  i2 = j; }
    }
    const int base = b * nc_per_b;
    idx[3 * (size_t)q + 0] = base + i0;
    idx[3 * (size_t)q + 1] = base + i1;
    idx[3 * (size_t)q + 2] = base + i2;
    wgt[3 * (size_t)q + 0] = 1.0f / fmaxf(d0, 1e-16f);
    wgt[3 * (size_t)q + 1] = 1.0f / fmaxf(d1, 1e-16f);
    wgt[3 * (size_t)q + 2] = 1.0f / fmaxf(d2, 1e-16f);
}

// ---------------------------------------------------------------------------
// k=1 gather + concat into padded GEMM input. grid.x = nq, grid.y over cols.
// ---------------------------------------------------------------------------
__global__ void gather1_concat(const float* __restrict__ feat, int Cf,
                               const int* __restrict__ idx,
                               const float* __restrict__ skip, int Cs,
                               float* __restrict__ out, int ldo)
{
    const int q = blockIdx.x;
    const int c = blockIdx.y * blockDim.x + threadIdx.x;
    if (c >= ldo) return;
    float v = 0.0f;
    if (c < Cf)            v = feat[(size_t)idx[q] * Cf + c];
    else if (c < Cf + Cs)  v = skip[(size_t)q * Cs + (c - Cf)];
    out[(size_t)q * ldo + c] = v;
}

// ---------------------------------------------------------------------------
// k=3 inverse-distance-weighted interpolation + concat (pads extra cols with 0).
// ---------------------------------------------------------------------------
__global__ void interp3_concat(const float* __restrict__ feat, int Cf,
                               const int* __restrict__ idx,
                               const float* __restrict__ wgt,
                               const float* __restrict__ skip, int Cs,
                               float* __restrict__ out, int ldo)
{
    const int q = blockIdx.x;
    const int c = blockIdx.y * blockDim.x + threadIdx.x;
    if (c >= ldo) return;
    float v = 0.0f;
    if (c < Cf) {
        const int   i0 = idx[3 * (size_t)q + 0];
        const int   i1 = idx[3 * (size_t)q + 1];
        const int   i2 = idx[3 * (size_t)q + 2];
        const float w0 = wgt[3 * (size_t)q + 0];
        const float w1 = wgt[3 * (size_t)q + 1];
        const float w2 = wgt[3 * (size_t)q + 2];
        v = (w0 * feat[(size_t)i0 * Cf + c] +
             w1 * feat[(size_t)i1 * Cf + c] +
             w2 * feat[(size_t)i2 * Cf + c]) / (w0 + w1 + w2);
    } else if (c < Cf + Cs) {
        v = skip[(size_t)q * Cs + (c - Cf)];
    }
    out[(size_t)q * ldo + c] = v;
}

// ---------------------------------------------------------------------------
// BatchNorm training-mode stats: one 256-thread block per channel.
// ---------------------------------------------------------------------------
__global__ __launch_bounds__(256) void bn_stats(const float* __restrict__ Y,
                                                int nrows, int C,
                                                float* __restrict__ mean,
                                                float* __restrict__ rsig)
{
    const int c = blockIdx.x;
    float s = 0.0f, s2 = 0.0f;
    for (int r = threadIdx.x; r < nrows; r += 256) {
        const float v = Y[(size_t)r * C + c];
        s += v; s2 += v * v;
    }
    __shared__ float sh[256], sh2[256];
    sh[threadIdx.x] = s; sh2[threadIdx.x] = s2;
    __syncthreads();
    for (int o = 128; o > 0; o >>= 1) {
        if (threadIdx.x < o) {
            sh[threadIdx.x]  += sh[threadIdx.x + o];
            sh2[threadIdx.x] += sh2[threadIdx.x + o];
        }
        __syncthreads();
    }
    if (threadIdx.x == 0) {
        const float m   = sh[0] / (float)nrows;
        const float var = sh2[0] / (float)nrows - m * m;
        mean[c] = m;
        rsig[c] = rsqrtf(var + 1e-5f);
    }
}

__global__ void bn_relu(float* __restrict__ Y, int C,
                        const float* __restrict__ mean,
                        const float* __restrict__ rsig,
                        const float* __restrict__ g,
                        const float* __restrict__ be)
{
    const int r = blockIdx.x;
    const int c = blockIdx.y * blockDim.x + threadIdx.x;
    if (c >= C) return;
    const size_t o = (size_t)r * C + c;
    const float v = (Y[o] - mean[c]) * rsig[c] * g[c] + be[c];
    Y[o] = fmaxf(v, 0.0f);
}

// Zero-padded weight copy: dst[drows,dcols] = src[rows,cols] padded with zeros.
__global__ void pad_weight(const float* __restrict__ src, int rows, int cols,
                           float* __restrict__ dst, int drows, int dcols)
{
    const int t = blockIdx.x * blockDim.x + threadIdx.x;
    if (t >= drows * dcols) return;
    const int r = t / dcols, c = t % dcols;
    dst[t] = (r < rows && c < cols) ? src[r * cols + c] : 0.0f;
}

// ---------------------------------------------------------------------------
// Host-side orchestration.
// ---------------------------------------------------------------------------
static inline void launch_gemm(const float* A, int lda, const float* W, int ldw,
                               const float* bias, float* C, int ldc,
                               int M, int N, int K, int nstore, int relu,
                               hipStream_t stream)
{
    dim3 grid(M / 256, (N + 63) / 64);
    gemm_wmma_f32<<<grid, 256, 0, stream>>>(A, lda, W, ldw, bias, C, ldc,
                                            M, N, K, nstore, relu);
}

extern "C" void kernel_launch(void* const* d_in, const int* in_sizes, int n_in,
                              void* d_out, int out_size, void* d_ws, size_t ws_size,
                              hipStream_t stream)
{
    // dict order, tuples flattened to leaves
    const float* x0   = (const float*)d_in[0];
    const float* pos0 = (const float*)d_in[1];
    const float* x1   = (const float*)d_in[2];
    const float* pos1 = (const float*)d_in[3];
    const float* x2   = (const float*)d_in[4];
    const float* pos2 = (const float*)d_in[5];
    const float* x3   = (const float*)d_in[6];
    const float* pos3 = (const float*)d_in[7];
    // d_in[8..11]: batch ids (contiguous equal-sized; implied by layout)
    const float* f3W1 = (const float*)d_in[12]; const float* f3b1 = (const float*)d_in[13];
    const float* f3g1 = (const float*)d_in[14]; const float* f3e1 = (const float*)d_in[15];
    const float* f3W2 = (const float*)d_in[16]; const float* f3b2 = (const float*)d_in[17];
    const float* f2W1 = (const float*)d_in[18]; const float* f2b1 = (const float*)d_in[19];
    const float* f2g1 = (const float*)d_in[20]; const float* f2e1 = (const float*)d_in[21];
    const float* f2W2 = (const float*)d_in[22]; const float* f2b2 = (const float*)d_in[23];
    const float* f1W1 = (const float*)d_in[24]; const float* f1b1 = (const float*)d_in[25];
    const float* f1g1 = (const float*)d_in[26]; const float* f1e1 = (const float*)d_in[27];
    const float* f1W2 = (const float*)d_in[28]; const float* f1b2 = (const float*)d_in[29];
    const float* f1g2 = (const float*)d_in[30]; const float* f1e2 = (const float*)d_in[31];
    const float* f1W3 = (const float*)d_in[32]; const float* f1b3 = (const float*)d_in[33];
    const float* hWa  = (const float*)d_in[34]; const float* hba  = (const float*)d_in[35];
    const float* hWb  = (const float*)d_in[36]; const float* hbb  = (const float*)d_in[37];
    const float* hWc  = (const float*)d_in[38]; const float* hbc  = (const float*)d_in[39];
    float* out = (float*)d_out;

    // workspace carve-up (floats)
    const size_t BUF = (size_t)N0T * 132;          // 8,650,752 floats per buffer
    float* ws   = (float*)d_ws;
    float* bufA = ws;
    float* bufB = bufA + BUF;
    float* bufC = bufB + BUF;
    float* W1p  = bufC + BUF;                      // padded fp1.W1 [132,128]
    float* Wcp  = W1p + 132 * 128;                 // padded head Wc [128,32]
    float* mean = Wcp + 128 * 32;
    float* rsig = mean + 256;
    float* wgt  = rsig + 256;                      // kNN weights [N0T*3]
    int*   idx  = (int*)(wgt + (size_t)N0T * 3);   // kNN indices [N0T*3]
    int*   idx1 = idx + (size_t)N0T * 3;           // k=1 indices [N2T]

    // ---------------- FP3: x3 [1024,1024] -> [4096,256] ----------------
    knn1_kernel<<<(N2T + 255) / 256, 256, 0, stream>>>(pos2, pos3, N2B, N3B, N2T, idx1);
    gather1_concat<<<dim3(N2T, (1280 + 255) / 256), 256, 0, stream>>>(
        x3, 1024, idx1, x2, 256, bufA, 1280);
    launch_gemm(bufA, 1280, f3W1, 256, f3b1, bufB, 256, N2T, 256, 1280, 256, 0, stream);
    bn_stats<<<256, 256, 0, stream>>>(bufB, N2T, 256, mean, rsig);
    bn_relu<<<dim3(N2T, 1), 256, 0, stream>>>(bufB, 256, mean, rsig, f3g1, f3e1);
    launch_gemm(bufB, 256, f3W2, 256, f3b2, bufA, 256, N2T, 256, 256, 256, 0, stream);
    // h_fp3 = bufA [4096,256]

    // ---------------- FP2: -> [16384,128] ----------------
    knn3_kernel<<<(N1T + 255) / 256, 256, 0, stream>>>(pos1, pos2, N1B, N2B, N1T, idx, wgt);
    interp3_concat<<<dim3(N1T, (384 + 255) / 256), 256, 0, stream>>>(
        bufA, 256, idx, wgt, x1, 128, bufB, 384);
    launch_gemm(bufB, 384, f2W1, 256, f2b1, bufC, 256, N1T, 256, 384, 256, 0, stream);
    bn_stats<<<256, 256, 0, stream>>>(bufC, N1T, 256, mean, rsig);
    bn_relu<<<dim3(N1T, 1), 256, 0, stream>>>(bufC, 256, mean, rsig, f2g1, f2e1);
    launch_gemm(bufC, 256, f2W2, 128, f2b2, bufA, 128, N1T, 128, 256, 128, 0, stream);
    // h_fp2 = bufA [16384,128]

    // ---------------- FP1: -> [65536,128] ----------------
    knn3_kernel<<<(N0T + 255) / 256, 256, 0, stream>>>(pos0, pos1, N0B, N1B, N0T, idx, wgt);
    pad_weight<<<(132 * 128 + 255) / 256, 256, 0, stream>>>(f1W1, 131, 128, W1p, 132, 128);
    interp3_concat<<<dim3(N0T, 1), 256, 0, stream>>>(
        bufA, 128, idx, wgt, x0, 3, bufB, 132);   // K padded 131 -> 132 (col 131 = 0)
    launch_gemm(bufB, 132, W1p, 128, f1b1, bufC, 128, N0T, 128, 132, 128, 0, stream);
    bn_stats<<<128, 256, 0, stream>>>(bufC, N0T, 128, mean, rsig);
    bn_relu<<<dim3(N0T, 1), 256, 0, stream>>>(bufC, 128, mean, rsig, f1g1, f1e1);
    launch_gemm(bufC, 128, f1W2, 128, f1b2, bufB, 128, N0T, 128, 128, 128, 0, stream);
    bn_stats<<<128, 256, 0, stream>>>(bufB, N0T, 128, mean, rsig);
    bn_relu<<<dim3(N0T, 1), 256, 0, stream>>>(bufB, 128, mean, rsig, f1g2, f1e2);
    launch_gemm(bufB, 128, f1W3, 128, f1b3, bufC, 128, N0T, 128, 128, 128, 0, stream);
    // h_fp1 = bufC [65536,128]

    // ---------------- Head: -> out [65536,3] ----------------
    launch_gemm(bufC, 128, hWa, 128, hba, bufB, 128, N0T, 128, 128, 128, 1, stream);
    launch_gemm(bufB, 128, hWb, 128, hbb, bufA, 128, N0T, 128, 128, 128, 1, stream);
    pad_weight<<<(128 * 32 + 255) / 256, 256, 0, stream>>>(hWc, 128, 3, Wcp, 128, 32);
    launch_gemm(bufA, 128, Wcp, 32, hbc, out, 3, N0T, 32, 128, 3, 0, stream);
}